// Loss_add_1322849927301
// MI455X (gfx1250) — compile-verified
//
#include <hip/hip_runtime.h>
#include <math.h>

// ---------------------------------------------------------------------------
// ADD-S loss on MI455X (gfx1250, wave32).
// d2[n,m] = qq[n] + rr[m] - 2 p.t ; min over m doesn't need qq -> add later.
// One V_WMMA_F32_16X16X4_F32 computes a full 16x16 tile of
// S[n,m] = rr[m] - 2 p.t  with A = [px,py,pz,1], B = [-2tx,-2ty,-2tz,rr].
// Each wave carries 3 N-tiles through one M-loop (3 WMMAs per B load), and
// the B operand is software-pipelined (prefetch mt+1 while computing mt).
// ---------------------------------------------------------------------------

typedef __attribute__((ext_vector_type(2))) float v2f;
typedef __attribute__((ext_vector_type(8))) float v8f;

#define NPOINTS   3000
#define BATCH     32
#define NBLK      8               // blocks per batch (split over N tiles)
#define MTILES    188             // ceil(3000/16); 188*16 = 3008
#define NT_BLK    24              // ceil(188/8) tiles per block
#define THREADS   256
#define WAVES     8               // wave32
#define BIGF      3.0e38f

// v_min_num_f32 without the fminf canonicalize (inputs are finite).
static __device__ __forceinline__ float fminfast(float a, float b) {
    float r;
    asm("v_min_num_f32 %0, %1, %2" : "=v"(r) : "v"(a), "v"(b));
    return r;
}

__global__ __launch_bounds__(THREADS)
void adds_loss_kernel(const float* __restrict__ pred_r,
                      const float* __restrict__ pred_t,
                      const float* __restrict__ target,
                      const float* __restrict__ model_points,
                      const int*   __restrict__ idx,
                      float*       __restrict__ partial)   // [BATCH*NBLK]
{
    __shared__ float4 sTgt[(MTILES + 1) * 16]; // {-2t, rr} + 1 pad tile  48384 B
    __shared__ float4 sPred[NT_BLK * 16];      // {px,py,pz,1}             6144 B
    __shared__ float  sQQ[NT_BLK * 16];        // |p|^2                    1536 B
    __shared__ float  sRed[WAVES];

    const int b   = blockIdx.x / NBLK;
    const int blk = blockIdx.x % NBLK;
    const int tid = threadIdx.x;

    // --- rotation from normalized quaternion (redundant per thread, cheap) ---
    float qa = pred_r[b * 4 + 0], qb = pred_r[b * 4 + 1];
    float qc = pred_r[b * 4 + 2], qd = pred_r[b * 4 + 3];
    float inv = rsqrtf(qa * qa + qb * qb + qc * qc + qd * qd);
    qa *= inv; qb *= inv; qc *= inv; qd *= inv;
    // R[k][j] per reference _quat_to_rot
    const float R00 = 1.f - 2.f * (qc * qc + qd * qd);
    const float R01 = 2.f * qb * qc - 2.f * qa * qd;
    const float R02 = 2.f * qa * qc + 2.f * qb * qd;
    const float R10 = 2.f * qb * qc + 2.f * qa * qd;
    const float R11 = 1.f - 2.f * (qb * qb + qd * qd);
    const float R12 = -2.f * qa * qb + 2.f * qc * qd;
    const float R20 = -2.f * qa * qc + 2.f * qb * qd;
    const float R21 = 2.f * qa * qb + 2.f * qc * qd;
    const float R22 = 1.f - 2.f * (qb * qb + qc * qc);
    const float tx = pred_t[b * 3 + 0];
    const float ty = pred_t[b * 3 + 1];
    const float tz = pred_t[b * 3 + 2];

    const int cls = idx[b];
    const bool is_sym = (cls == 0) | (cls == 3) | (cls == 7) | (cls == 10);

    const float* __restrict__ tb = target       + (size_t)b * NPOINTS * 3;
    const float* __restrict__ mb = model_points + (size_t)b * NPOINTS * 3;

    // --- stage targets: [-2t, rr]; pad (incl. prefetch tile) with rr=+huge ---
    for (int m = tid; m < (MTILES + 1) * 16; m += THREADS) {
        float4 v;
        if (m < NPOINTS) {
            float x = tb[m * 3 + 0], y = tb[m * 3 + 1], z = tb[m * 3 + 2];
            v.x = -2.f * x; v.y = -2.f * y; v.z = -2.f * z;
            v.w = x * x + y * y + z * z;
        } else {
            v.x = 0.f; v.y = 0.f; v.z = 0.f; v.w = BIGF;
        }
        sTgt[m] = v;
    }

    // --- stage this block's pred slice: [p, 1] and qq ---
    const int n0 = blk * NT_BLK * 16;
    for (int i = tid; i < NT_BLK * 16; i += THREADS) {
        const int n = n0 + i;
        float4 v; float qq = 0.f;
        if (n < NPOINTS) {
            float mx = mb[n * 3 + 0], my = mb[n * 3 + 1], mz = mb[n * 3 + 2];
            float px = mx * R00 + my * R10 + mz * R20 + tx;
            float py = mx * R01 + my * R11 + mz * R21 + ty;
            float pz = mx * R02 + my * R12 + mz * R22 + tz;
            v.x = px; v.y = py; v.z = pz; v.w = 1.f;
            qq = px * px + py * py + pz * pz;
        } else {
            v.x = 0.f; v.y = 0.f; v.z = 0.f; v.w = 1.f;
        }
        sPred[i] = v;
        sQQ[i]   = qq;
    }
    __syncthreads();

    const int wave = tid >> 5;
    const int lane = tid & 31;
    const int tilesThisBlk = min(NT_BLK, MTILES - blk * NT_BLK); // last blk: 20
    float waveSum = 0.f;

    if (is_sym) {
        // A/B operand addressing (identical for both): row/col = lane&15,
        // K pair = (lane>>4)*2 (ISA 7.12.2: 32-bit A 16x4 layout; B mirrors).
        const int rc = lane & 15;
        const int kb = (lane >> 4) << 1;

        // This wave's three N-tiles; invalid ones alias tile t0 (uniform EXEC
        // for WMMA) and are skipped at accumulation.
        const int  t0 = wave, t1 = wave + WAVES, t2 = wave + 2 * WAVES;
        const bool val1 = (t1 < tilesThisBlk);
        const bool val2 = (t2 < tilesThisBlk);
        const int  u1 = val1 ? t1 : t0;
        const int  u2 = val2 ? t2 : t0;

        const float* a0p = (const float*)&sPred[t0 * 16 + rc];
        const float* a1p = (const float*)&sPred[u1 * 16 + rc];
        const float* a2p = (const float*)&sPred[u2 * 16 + rc];
        v2f A0; A0.x = a0p[kb]; A0.y = a0p[kb + 1];
        v2f A1; A1.x = a1p[kb]; A1.y = a1p[kb + 1];
        v2f A2; A2.x = a2p[kb]; A2.y = a2p[kb + 1];

        v8f mn0, mn1, mn2;
#pragma unroll
        for (int v = 0; v < 8; ++v) { mn0[v] = BIGF; mn1[v] = BIGF; mn2[v] = BIGF; }

        // software-pipelined B operand: prefetch tile mt+1 while computing mt
        const float* bp0 = (const float*)&sTgt[0 * 16 + rc];
        v2f Bc; Bc.x = bp0[kb]; Bc.y = bp0[kb + 1];

#pragma unroll 2
        for (int mt = 0; mt < MTILES; ++mt) {
            const float* bpn = (const float*)&sTgt[(mt + 1) * 16 + rc]; // padded
            v2f Bn; Bn.x = bpn[kb]; Bn.y = bpn[kb + 1];

            v8f c = {};
            v8f d0 = __builtin_amdgcn_wmma_f32_16x16x4_f32(
                false, A0, false, Bc, (short)0, c, false, false);
            v8f d1 = __builtin_amdgcn_wmma_f32_16x16x4_f32(
                false, A1, false, Bc, (short)0, c, false, false);
            v8f d2 = __builtin_amdgcn_wmma_f32_16x16x4_f32(
                false, A2, false, Bc, (short)0, c, false, false);
#pragma unroll
            for (int v = 0; v < 8; ++v) {
                mn0[v] = fminfast(mn0[v], d0[v]);
                mn1[v] = fminfast(mn1[v], d1[v]);
                mn2[v] = fminfast(mn2[v], d2[v]);
            }
            Bc = Bn;
        }

        // Per-tile finalize: min across 16 columns (xor 1..8 stays inside each
        // 16-lane half), then lanes 0/16 own rows v / v+8.
        v8f mins[3] = { mn0, mn1, mn2 };
        const int  ts[3]  = { t0, u1, u2 };
        const bool vs[3]  = { true, val1, val2 };
#pragma unroll
        for (int k = 0; k < 3; ++k) {
            v8f mn = mins[k];
#pragma unroll
            for (int off = 1; off <= 8; off <<= 1) {
#pragma unroll
                for (int v = 0; v < 8; ++v)
                    mn[v] = fminfast(mn[v], __shfl_xor(mn[v], off, 32));
            }
            if (vs[k] && (lane & 15) == 0) {
                const int rbase = ts[k] * 16 + ((lane >> 4) << 3);
                float s = 0.f;
#pragma unroll
                for (int v = 0; v < 8; ++v) {
                    const int nloc = rbase + v;
                    if (n0 + nloc < NPOINTS) {
                        float d2 = sQQ[nloc] + mn[v];
                        s += sqrtf(fmaxf(d2, 0.f));
                    }
                }
                waveSum += s;
            }
        }
        // fold lane16 partial into lane0
        waveSum += __shfl_xor(waveSum, 16, 32);
    } else {
        // non-symmetric: dis_n = ||p_n - t_n||
        for (int i = tid; i < tilesThisBlk * 16; i += THREADS) {
            const int n = n0 + i;
            if (n < NPOINTS) {
                float4 p = sPred[i];
                float dx = p.x - tb[n * 3 + 0];
                float dy = p.y - tb[n * 3 + 1];
                float dz = p.z - tb[n * 3 + 2];
                waveSum += sqrtf(dx * dx + dy * dy + dz * dz);
            }
        }
#pragma unroll
        for (int off = 1; off < 32; off <<= 1)
            waveSum += __shfl_xor(waveSum, off, 32);
    }

    if (lane == 0) sRed[wave] = waveSum;
    __syncthreads();
    if (tid == 0) {
        float s = 0.f;
#pragma unroll
        for (int w = 0; w < WAVES; ++w) s += sRed[w];
        partial[b * NBLK + blk] = s;
    }
}

__global__ void adds_finalize_kernel(const float* __restrict__ partial,
                                     float* __restrict__ out)
{
    const int b = threadIdx.x;
    if (b < BATCH) {
        float s = 0.f;
#pragma unroll
        for (int i = 0; i < NBLK; ++i) s += partial[b * NBLK + i];
        out[b] = s * (1.0f / (float)NPOINTS);
    }
}

extern "C" void kernel_launch(void* const* d_in, const int* in_sizes, int n_in,
                              void* d_out, int out_size, void* d_ws, size_t ws_size,
                              hipStream_t stream)
{
    const float* pred_r       = (const float*)d_in[0];
    const float* pred_t       = (const float*)d_in[1];
    const float* target       = (const float*)d_in[2];
    const float* model_points = (const float*)d_in[3];
    const int*   idx          = (const int*)d_in[4];
    float* out     = (float*)d_out;
    float* partial = (float*)d_ws;   // BATCH*NBLK floats

    adds_loss_kernel<<<dim3(BATCH * NBLK), dim3(THREADS), 0, stream>>>(
        pred_r, pred_t, target, model_points, idx, partial);
    adds_finalize_kernel<<<1, 32, 0, stream>>>(partial, out);
}